// SAGE_6571299963286
// MI455X (gfx1250) — compile-verified
//
#include <hip/hip_runtime.h>
#include <hip/hip_bf16.h>

typedef __attribute__((ext_vector_type(2))) float v2f;
typedef __attribute__((ext_vector_type(8))) float v8f;

#define DFEAT 128

// ---------------------------------------------------------------------------
// Zero a float buffer (msg accumulator + degree counter, contiguous).
// float4 stores; caller guarantees 16B alignment and a multiple-of-4 count
// (the float count is rounded up to whole float4s; no scalar tail).
// ---------------------------------------------------------------------------
__global__ void sage_zero_kernel(float4* __restrict__ p, size_t n4) {
    size_t i = (size_t)blockIdx.x * blockDim.x + threadIdx.x;
    if (i < n4) p[i] = make_float4(0.f, 0.f, 0.f, 0.f);
}

// ---------------------------------------------------------------------------
// Edge scatter: one wave32 per edge. Each lane handles 4 features (float4
// gather from the source row -> 4 global_atomic_add_f32 into the dst row).
// Lane 0 bumps the degree counter. Node features (25.6MB) are resident in the
// 192MB L2, so both the gather and the atomic RMWs are serviced by L2.
// ---------------------------------------------------------------------------
__global__ __launch_bounds__(256)
void sage_scatter_kernel(const float* __restrict__ h,
                         const int* __restrict__ src,
                         const int* __restrict__ dst,
                         float* __restrict__ msg,
                         float* __restrict__ deg,
                         int n_edges) {
    int edge = (int)((blockIdx.x * (size_t)blockDim.x + threadIdx.x) >> 5);
    int lane = threadIdx.x & 31;
    if (edge >= n_edges) return;
    int s = src[edge];
    int d = dst[edge];
    const float4* hs = reinterpret_cast<const float4*>(h + (size_t)s * DFEAT);
    float4 v = hs[lane];
    float* mp = msg + (size_t)d * DFEAT + lane * 4;
    atomicAdd(mp + 0, v.x);
    atomicAdd(mp + 1, v.y);
    atomicAdd(mp + 2, v.z);
    atomicAdd(mp + 3, v.w);
    if (lane == 0) atomicAdd(deg + d, 1.0f);
}

// ---------------------------------------------------------------------------
// Fused SAGE layer: out = (h @ W_self) + ((msg/max(deg,1)) @ W_neigh) + b,
// optional compile-time ReLU. 256 threads = 8 waves; block covers 16 rows x
// 128 cols. All waves share the 16x128 A tiles (LDS, padded to 132 words for
// bank-conflict-free column reads); wave w computes output columns
// [16w, 16w+16) with exact-precision V_WMMA_F32_16X16X4_F32, accumulating
// both GEMMs into one 16x16 fp32 C tile (8 VGPRs/lane).
//
// f32 16x16x4 per-lane layout (ISA 7.12.2):
//   A: lanes 0-15 -> M=lane,    v0=K(k0),  v1=K(k0+1)
//      lanes16-31 -> M=lane-16, v0=K(k0+2),v1=K(k0+3)
//   B: same K split; N = lane&15.
//   C: VGPR v holds row M=v (lanes 0-15) / M=v+8 (lanes 16-31), N=lane&15.
// ---------------------------------------------------------------------------
template <int RELU>
__global__ __launch_bounds__(256)
void sage_gemm_kernel(const float* __restrict__ h,
                      const float* __restrict__ msg,
                      const float* __restrict__ deg,
                      const float* __restrict__ Wself,
                      const float* __restrict__ Wneigh,
                      const float* __restrict__ bias,
                      float* __restrict__ out,
                      int n_nodes) {
    __shared__ float As[16][132];
    __shared__ float An[16][132];

    const int m0 = blockIdx.x * 16;
    const bool full = (m0 + 16) <= n_nodes;  // block-uniform: no lane divergence

    // Cooperative stage of the 16x128 self & (mean-normalized) neighbor tiles.
    if (full) {
        for (int idx = threadIdx.x; idx < 16 * DFEAT; idx += 256) {
            int r = idx >> 7, c = idx & 127;
            int row = m0 + r;
            float dg = fmaxf(deg[row], 1.0f);
            As[r][c] = h[(size_t)row * DFEAT + c];
            An[r][c] = msg[(size_t)row * DFEAT + c] / dg;
        }
    } else {
        for (int idx = threadIdx.x; idx < 16 * DFEAT; idx += 256) {
            int r = idx >> 7, c = idx & 127;
            int row = m0 + r;
            float hs = 0.0f, hn = 0.0f;
            if (row < n_nodes) {
                hs = h[(size_t)row * DFEAT + c];
                float dg = fmaxf(deg[row], 1.0f);
                hn = msg[(size_t)row * DFEAT + c] / dg;
            }
            As[r][c] = hs;
            An[r][c] = hn;
        }
    }
    __syncthreads();

    const int lane = threadIdx.x & 31;
    const int wave = threadIdx.x >> 5;
    const int n0 = wave * 16;
    const int M = lane & 15;
    const int half = lane >> 4;

    v8f acc = {};

#pragma unroll 4
    for (int k0 = 0; k0 < DFEAT; k0 += 4) {
        const int kb = k0 + 2 * half;
        v2f a_s, a_n, b_s, b_n;
        a_s.x = As[M][kb];
        a_s.y = As[M][kb + 1];
        a_n.x = An[M][kb];
        a_n.y = An[M][kb + 1];
        b_s.x = Wself[(size_t)kb * DFEAT + n0 + M];
        b_s.y = Wself[(size_t)(kb + 1) * DFEAT + n0 + M];
        b_n.x = Wneigh[(size_t)kb * DFEAT + n0 + M];
        b_n.y = Wneigh[(size_t)(kb + 1) * DFEAT + n0 + M];
        acc = __builtin_amdgcn_wmma_f32_16x16x4_f32(
            false, a_s, false, b_s, (short)0, acc, false, false);
        acc = __builtin_amdgcn_wmma_f32_16x16x4_f32(
            false, a_n, false, b_n, (short)0, acc, false, false);
    }

    const float bv = bias[n0 + M];
    float* obase = out + (size_t)(m0 + 8 * half) * DFEAT + n0 + M;

    if (full) {
        // Straight-line epilogue: no exec-mask churn.
#pragma unroll
        for (int v = 0; v < 8; ++v) {
            float val = acc[v] + bv;
            if (RELU) val = fmaxf(val, 0.0f);
            obase[(size_t)v * DFEAT] = val;
        }
    } else {
#pragma unroll
        for (int v = 0; v < 8; ++v) {
            int row = m0 + v + 8 * half;
            if (row < n_nodes) {
                float val = acc[v] + bv;
                if (RELU) val = fmaxf(val, 0.0f);
                out[(size_t)row * DFEAT + n0 + M] = val;
            }
        }
    }
}

// ---------------------------------------------------------------------------
// Driver: layer1 (scatter -> fused GEMM + ReLU into h1), layer2 (scatter over
// h1 -> fused GEMM into d_out). All launches on `stream`; workspace layout:
//   [ msg: N*128 | deg: N | h1: N*128 ]  (floats)
// ---------------------------------------------------------------------------
extern "C" void kernel_launch(void* const* d_in, const int* in_sizes, int n_in,
                              void* d_out, int out_size, void* d_ws, size_t ws_size,
                              hipStream_t stream) {
    const float* x   = (const float*)d_in[0];
    const float* W1s = (const float*)d_in[1];
    const float* W1n = (const float*)d_in[2];
    const float* b1  = (const float*)d_in[3];
    const float* W2s = (const float*)d_in[4];
    const float* W2n = (const float*)d_in[5];
    const float* b2  = (const float*)d_in[6];
    const int*   src = (const int*)d_in[7];
    const int*   dst = (const int*)d_in[8];

    const int n_nodes = in_sizes[0] / DFEAT;
    const int n_edges = in_sizes[7];

    float* msg = (float*)d_ws;
    float* deg = msg + (size_t)n_nodes * DFEAT;
    float* h1  = deg + n_nodes;
    float* outp = (float*)d_out;

    // msg (N*128 floats) + deg (N floats) are contiguous: N*129 floats total.
    // N*129 may not be a multiple of 4; round UP in float4 units — the extra
    // <=3 floats land in the h1 region, rewritten by the layer-1 GEMM, then clobbered by the layer-2 zero.
    const size_t zf   = (size_t)n_nodes * (DFEAT + 1);
    const size_t z4   = (zf + 3) / 4;
    const int zblocks = (int)((z4 + 255) / 256);
    const int sblocks = (n_edges + 7) / 8;   // 8 waves (edges) per block
    const int gblocks = (n_nodes + 15) / 16;

    // Layer 1
    sage_zero_kernel<<<zblocks, 256, 0, stream>>>((float4*)msg, z4);
    sage_scatter_kernel<<<sblocks, 256, 0, stream>>>(x, src, dst, msg, deg, n_edges);
    sage_gemm_kernel<1><<<gblocks, 256, 0, stream>>>(x, msg, deg, W1s, W1n, b1, h1,
                                                     n_nodes);
    // Layer 2
    sage_zero_kernel<<<zblocks, 256, 0, stream>>>((float4*)msg, z4);
    sage_scatter_kernel<<<sblocks, 256, 0, stream>>>(h1, src, dst, msg, deg, n_edges);
    sage_gemm_kernel<0><<<gblocks, 256, 0, stream>>>(h1, msg, deg, W2s, W2n, b2, outp,
                                                     n_nodes);
}